// DCCF_81793357185705
// MI455X (gfx1250) — compile-verified
//
#include <hip/hip_runtime.h>
#include <hip/hip_bf16.h>
#include <stdint.h>

// ---------------- problem constants (match reference) ----------------
#define NUM_USER  50000
#define NUM_ITEM  25000
#define NN        (NUM_USER + NUM_ITEM)   // 75000
#define DIM       64
#define NINT      128
#define NE        800000
#define NLAYERS   2

typedef __attribute__((ext_vector_type(2))) float v2f;
typedef __attribute__((ext_vector_type(8))) float v8f;

// =====================================================================
// init: x = concat(user_emb, item_emb) (float4 / B128); deg = 0
// i4 indexes 16-byte chunks; 16 chunks per 64-dim row.
// =====================================================================
__global__ void __launch_bounds__(256)
init_kernel(const float4* __restrict__ ue, const float4* __restrict__ ie,
            float4* __restrict__ x, float* __restrict__ deg)
{
    int i = blockIdx.x * blockDim.x + threadIdx.x;
    if (i < NN * (DIM / 4)) {
        int node = i >> 4;
        x[i] = (node < NUM_USER) ? ue[i] : ie[i - NUM_USER * (DIM / 4)];
    }
    if (i < NN) deg[i] = 0.0f;
}

__global__ void __launch_bounds__(256)
zero4_kernel(float4* __restrict__ p, int n4)
{
    int i = blockIdx.x * blockDim.x + threadIdx.x;
    if (i < n4) p[i] = make_float4(0.f, 0.f, 0.f, 0.f);
}

// =====================================================================
// degree: deg[h] += 1, deg[t] += 1 per edge
// =====================================================================
__global__ void __launch_bounds__(256)
degree_kernel(const int* __restrict__ h_idx, const int* __restrict__ t_idx,
              float* __restrict__ deg)
{
    int e = blockIdx.x * blockDim.x + threadIdx.x;
    if (e >= NE) return;
    atomicAdd(&deg[h_idx[e]], 1.0f);
    atomicAdd(&deg[t_idx[e]], 1.0f);
}

// deg -> rsqrt(deg + 1e-7) in place
__global__ void __launch_bounds__(256)
dinv_kernel(float* __restrict__ deg)
{
    int i = blockIdx.x * blockDim.x + threadIdx.x;
    if (i < NN) deg[i] = rsqrtf(deg[i] + 1e-7f);
}

// w_edge[e] = dinv[h]*dinv[t]
__global__ void __launch_bounds__(256)
wedge_kernel(const int* __restrict__ h_idx, const int* __restrict__ t_idx,
             const float* __restrict__ dinv, float* __restrict__ w)
{
    int e = blockIdx.x * blockDim.x + threadIdx.x;
    if (e < NE) w[e] = dinv[h_idx[e]] * dinv[t_idx[e]];
}

// =====================================================================
// symmetric SpMM: out[h] += w*x[t]; out[t] += w*x[h]   (thread = edge,dim)
// dims are the fast index -> gathers x[node*64 + d] are coalesced per wave;
// node table (19.2 MB) is fully L2-resident on MI455X (192 MB L2).
// =====================================================================
__global__ void __launch_bounds__(256)
spmm_sym_kernel(const int* __restrict__ h_idx, const int* __restrict__ t_idx,
                const float* __restrict__ w, const float* __restrict__ x,
                float* __restrict__ out)
{
    long idx = (long)blockIdx.x * blockDim.x + threadIdx.x;
    if (idx >= (long)NE * DIM) return;
    int e = (int)(idx >> 6);
    int d = (int)(idx & 63);
    int h = h_idx[e], t = t_idx[e];
    float we = w[e];
    atomicAdd(&out[h * DIM + d], we * x[t * DIM + d]);
    atomicAdd(&out[t * DIM + d], we * x[h * DIM + d]);
}

// directed adaptive SpMM: out[h] += alpha[e] * x[t]
__global__ void __launch_bounds__(256)
spmm_adapt_kernel(const int* __restrict__ h_idx, const int* __restrict__ t_idx,
                  const float* __restrict__ alpha, const float* __restrict__ x,
                  float* __restrict__ out)
{
    long idx = (long)blockIdx.x * blockDim.x + threadIdx.x;
    if (idx >= (long)NE * DIM) return;
    int e = (int)(idx >> 6);
    int d = (int)(idx & 63);
    atomicAdd(&out[h_idx[e] * DIM + d], alpha[e] * x[t_idx[e] * DIM + d]);
}

// =====================================================================
// alpha[e] = (cos(emb[h], emb[t]) + 1) * 0.5  — one wave32 per edge,
// lane-parallel over dims, __shfl_xor tree reduction (wave32 native).
// =====================================================================
__global__ void __launch_bounds__(256)
alpha_kernel(const int* __restrict__ h_idx, const int* __restrict__ t_idx,
             const float* __restrict__ emb, float* __restrict__ alpha)
{
    int gtid = blockIdx.x * blockDim.x + threadIdx.x;
    int e    = gtid >> 5;
    int lane = threadIdx.x & 31;
    if (e >= NE) return;
    int h = h_idx[e], t = t_idx[e];
    float h0 = emb[h * DIM + lane], h1 = emb[h * DIM + 32 + lane];
    float t0 = emb[t * DIM + lane], t1 = emb[t * DIM + 32 + lane];
    float dot = h0 * t0 + h1 * t1;
    float nh  = h0 * h0 + h1 * h1;
    float nt  = t0 * t0 + t1 * t1;
    #pragma unroll
    for (int off = 16; off > 0; off >>= 1) {
        dot += __shfl_xor(dot, off, 32);
        nh  += __shfl_xor(nh,  off, 32);
        nt  += __shfl_xor(nt,  off, 32);
    }
    if (lane == 0) {
        float dn = fmaxf(sqrtf(nh), 1e-12f) * fmaxf(sqrtf(nt), 1e-12f);
        alpha[e] = (dot / dn + 1.0f) * 0.5f;
    }
}

// =====================================================================
// Intent projection with fp32 WMMA:
//   out = softmax(emb @ W, axis=1) @ W^T      emb: n x 64, W: 64 x 128
// Block = 64 threads (2 waves); each wave owns a 16-row node tile.
//   stage W -> LDS via GLOBAL_LOAD_ASYNC_TO_LDS_B128 (ASYNCcnt path)
//   GEMM1: A(16x64) x W(64x128)  -> 16 K-steps x 8 N-tiles of wmma 16x16x4 f32
//   softmax in LDS (per-wave 16x128 tile, stride-132 padded)
//   GEMM2: P(16x128) x W^T(128x64) -> 32 K-steps x 4 N-tiles
// W row stride 132 floats (528B = 33*16B): keeps 16B alignment for the B128
// async LDS writes and is conflict-free on 64 banks for BOTH access patterns:
//   GEMM1  Ws[ka*132 + n]  (consecutive n across lanes)
//   GEMM2  Ws[n*132 + ka]  (bank = (4n + ka) mod 64, distinct per lane group)
// =====================================================================
#define WS 132          // LDS row stride for W  (64 rows x 128 cols)
#define LS 132          // LDS row stride for logits (16 rows x 128 cols)

__global__ void __launch_bounds__(64)
intent_kernel(const float* __restrict__ emb, const float* __restrict__ Wg,
              float* __restrict__ out, int n_nodes)
{
    __shared__ __align__(16) float Ws[DIM * WS];   // 33792 B
    __shared__ __align__(16) float Lg[2][16 * LS]; // 16896 B (per-wave logits)

    const int tid  = threadIdx.x;
    const int wave = tid >> 5;
    const int lane = tid & 31;
    const int mlo  = lane & 15;
    const bool hi  = lane >= 16;

    // ---- async stage of W into LDS: one B128 per lane per row --------------
    // lane l of the issuing wave moves row bytes [16l, 16l+16) directly to LDS.
    {
        unsigned lds_base = (unsigned)(uintptr_t)(&Ws[0]);
        for (int r = wave; r < DIM; r += 2) {
            unsigned lds = lds_base + (unsigned)(r * (WS * 4)) + (unsigned)(lane * 16);
            unsigned long long ga =
                (unsigned long long)(uintptr_t)(Wg + r * NINT) + (unsigned)(lane * 16);
            asm volatile("global_load_async_to_lds_b128 %0, %1, off"
                         :: "v"(lds), "v"(ga) : "memory");
        }
        asm volatile("s_wait_asynccnt 0x0" ::: "memory");
    }
    __syncthreads();

    const int row0 = (blockIdx.x * 2 + wave) * 16;
    if (row0 >= n_nodes) return;          // whole-wave exit: EXEC stays all-1 for WMMA

    int mrow = row0 + mlo;                // clamp tail reads; tail rows never stored
    if (mrow >= n_nodes) mrow = n_nodes - 1;
    const float* erow = emb + (size_t)mrow * DIM;
    float* Lrow = &Lg[wave][0];

    // ---------------- GEMM1: logits = E @ W ----------------
    v8f acc[8] = {};
    #pragma unroll
    for (int kk = 0; kk < DIM; kk += 4) {
        const int ka = kk + (hi ? 2 : 0);     // A: lanes 16-31 hold K+2,K+3
        v2f a;
        a.x = erow[ka];
        a.y = erow[ka + 1];
        #pragma unroll
        for (int t = 0; t < 8; ++t) {
            const int n = t * 16 + mlo;       // B: lane -> column N
            v2f b;
            b.x = Ws[ka * WS + n];            // row K (or K+2 in hi half)
            b.y = Ws[(ka + 1) * WS + n];      // row K+1 (or K+3)
            acc[t] = __builtin_amdgcn_wmma_f32_16x16x4_f32(
                false, a, false, b, (short)0, acc[t], false, false);
        }
    }

    // spill logits: C layout -> lane holds column N, VGPR v holds row v (+8 hi)
    #pragma unroll
    for (int t = 0; t < 8; ++t)
        #pragma unroll
        for (int v = 0; v < 8; ++v) {
            int m = v + (hi ? 8 : 0);
            Lrow[m * LS + t * 16 + mlo] = acc[t][v];
        }
    // same-wave LDS ops are in order (DScnt) — no cross-wave barrier needed.

    // ---------------- softmax over 128 cols, 2 lanes per row ----------------
    {
        float* r = &Lrow[mlo * LS + (hi ? 64 : 0)];
        float mx = -3.402823466e38f;
        for (int c = 0; c < 64; ++c) mx = fmaxf(mx, r[c]);
        mx = fmaxf(mx, __shfl_xor(mx, 16, 32));
        float s = 0.0f;
        for (int c = 0; c < 64; ++c) { float ev = __expf(r[c] - mx); r[c] = ev; s += ev; }
        s += __shfl_xor(s, 16, 32);
        float inv = 1.0f / s;
        for (int c = 0; c < 64; ++c) r[c] *= inv;
    }

    // ---------------- GEMM2: out = P @ W^T ----------------
    v8f acc2[4] = {};
    #pragma unroll
    for (int kk = 0; kk < NINT; kk += 4) {
        const int ka = kk + (hi ? 2 : 0);
        v2f a;
        a.x = Lrow[mlo * LS + ka];
        a.y = Lrow[mlo * LS + ka + 1];
        #pragma unroll
        for (int t = 0; t < 4; ++t) {
            const int n = t * 16 + mlo;       // output dim
            v2f b;                            // B2[k][n] = W[n][k]
            b.x = Ws[n * WS + ka];
            b.y = Ws[n * WS + ka + 1];
            acc2[t] = __builtin_amdgcn_wmma_f32_16x16x4_f32(
                false, a, false, b, (short)0, acc2[t], false, false);
        }
    }

    // store: per VGPR v, lanes of one half write one node row, contiguous dims
    #pragma unroll
    for (int t = 0; t < 4; ++t)
        #pragma unroll
        for (int v = 0; v < 8; ++v) {
            int m = v + (hi ? 8 : 0);
            int node = row0 + m;
            if (node < n_nodes)
                out[(size_t)node * DIM + t * 16 + mlo] = acc2[t][v];
        }
}

// =====================================================================
// x += gnn + intl + gaa + iaa   (float4 / B128 streams)
// =====================================================================
__global__ void __launch_bounds__(256)
combine_kernel(float4* __restrict__ x, const float4* __restrict__ g,
               const float4* __restrict__ it, const float4* __restrict__ ga,
               const float4* __restrict__ ia)
{
    int i = blockIdx.x * blockDim.x + threadIdx.x;
    if (i >= NN * (DIM / 4)) return;
    float4 a = g[i], b = it[i], c = ga[i], d = ia[i], e = x[i];
    x[i] = make_float4(a.x + b.x + c.x + d.x + e.x,
                       a.y + b.y + c.y + d.y + e.y,
                       a.z + b.z + c.z + d.z + e.z,
                       a.w + b.w + c.w + d.w + e.w);
}

// =====================================================================
static inline int cdiv(long a, long b) { return (int)((a + b - 1) / b); }

extern "C" void kernel_launch(void* const* d_in, const int* in_sizes, int n_in,
                              void* d_out, int out_size, void* d_ws, size_t ws_size,
                              hipStream_t stream)
{
    const float* user_emb = (const float*)d_in[0];
    const float* item_emb = (const float*)d_in[1];
    const float* user_int = (const float*)d_in[2];
    const float* item_int = (const float*)d_in[3];
    const int*   h_idx    = (const int*)d_in[4];
    const int*   t_idx    = (const int*)d_in[5];
    float* out = (float*)d_out;

    // workspace layout (floats): x | deg/dinv | w_edge | alpha
    float* x     = (float*)d_ws;
    float* deg   = x + (size_t)NN * DIM;
    float* wedge = deg + NN;
    float* alpha = wedge + NE;

    const int B = 256;
    const size_t R  = (size_t)NN * DIM;
    const int    R4 = NN * (DIM / 4);

    init_kernel  <<<cdiv(R4, B), B, 0, stream>>>((const float4*)user_emb,
                                                 (const float4*)item_emb,
                                                 (float4*)x, deg);
    degree_kernel<<<cdiv(NE, B), B, 0, stream>>>(h_idx, t_idx, deg);
    dinv_kernel  <<<cdiv(NN, B), B, 0, stream>>>(deg);
    wedge_kernel <<<cdiv(NE, B), B, 0, stream>>>(h_idx, t_idx, deg, wedge);

    for (int l = 0; l < NLAYERS; ++l) {
        float* gnn  = out + (size_t)(0 * NLAYERS + l) * R;
        float* intl = out + (size_t)(1 * NLAYERS + l) * R;
        float* gaa  = out + (size_t)(2 * NLAYERS + l) * R;
        float* iaa  = out + (size_t)(3 * NLAYERS + l) * R;

        zero4_kernel<<<cdiv(R4, B), B, 0, stream>>>((float4*)gnn, R4);
        zero4_kernel<<<cdiv(R4, B), B, 0, stream>>>((float4*)gaa, R4);
        zero4_kernel<<<cdiv(R4, B), B, 0, stream>>>((float4*)iaa, R4);

        // gnn = normalized-adjacency SpMM
        spmm_sym_kernel<<<cdiv((long)NE * DIM, B), B, 0, stream>>>(
            h_idx, t_idx, wedge, x, gnn);

        // intl = softmax(E @ W) @ W^T  (WMMA)
        intent_kernel<<<cdiv(NUM_USER, 32), 64, 0, stream>>>(
            x, user_int, intl, NUM_USER);
        intent_kernel<<<cdiv(NUM_ITEM, 32), 64, 0, stream>>>(
            x + (size_t)NUM_USER * DIM, item_int, intl + (size_t)NUM_USER * DIM,
            NUM_ITEM);

        // gaa = adaptive(gnn):  alpha from gnn rows, scatter alpha*x[t] -> h
        alpha_kernel<<<cdiv((long)NE * 32, B), B, 0, stream>>>(h_idx, t_idx, gnn, alpha);
        spmm_adapt_kernel<<<cdiv((long)NE * DIM, B), B, 0, stream>>>(
            h_idx, t_idx, alpha, x, gaa);

        // iaa = adaptive(intl)
        alpha_kernel<<<cdiv((long)NE * 32, B), B, 0, stream>>>(h_idx, t_idx, intl, alpha);
        spmm_adapt_kernel<<<cdiv((long)NE * DIM, B), B, 0, stream>>>(
            h_idx, t_idx, alpha, x, iaa);

        // x = gnn + intl + gaa + iaa + x
        combine_kernel<<<cdiv(R4, B), B, 0, stream>>>(
            (float4*)x, (const float4*)gnn, (const float4*)intl,
            (const float4*)gaa, (const float4*)iaa);
    }
}